// GraphNeuralEmbedder_45131516346425
// MI455X (gfx1250) — compile-verified
//
#include <hip/hip_runtime.h>
#include <hip/hip_bf16.h>

typedef float v2f __attribute__((ext_vector_type(2)));
typedef float v8f __attribute__((ext_vector_type(8)));

// ---------------------------------------------------------------------------
// Degree / normalization
// ---------------------------------------------------------------------------
__global__ void deg_kernel(const int* __restrict__ dst, float* __restrict__ deg, int E) {
    int e = blockIdx.x * blockDim.x + threadIdx.x;
    if (e < E) atomicAdd(&deg[dst[e]], 1.0f);
}

__global__ void dinv_kernel(float* __restrict__ deg, int N) {
    int i = blockIdx.x * blockDim.x + threadIdx.x;
    if (i < N) deg[i] = rsqrtf(deg[i] + 1.0f);   // deg includes self-loop
}

__global__ void edge_norm_kernel(const int* __restrict__ src, const int* __restrict__ dst,
                                 const float* __restrict__ dinv, float* __restrict__ norm, int E) {
    int e = blockIdx.x * blockDim.x + threadIdx.x;
    if (e < E) norm[e] = dinv[src[e]] * dinv[dst[e]];
}

// ---------------------------------------------------------------------------
// Dense transform  C[M, NT*16] = A[M,K] @ W[K, NT*16]  via V_WMMA_F32_16X16X4_F32.
// One wave32 per 16-row strip; NT independent accumulators cover the whole N
// dimension so (a) the A slice is loaded once per k-step and reused NT times,
// and (b) the NT WMMAs per k-step are independent -> back-to-back issue into
// the matrix pipe instead of a serial RAW chain on one accumulator.
//   A (16x4 f32): lane r=lane&15 is row M=r; group g=lane>>4 holds K={2g,2g+1}.
//   B (4x16 f32): vgpr j, group g holds row K=j+2g, col N=r.
//   C (16x16 f32): vgpr i, group g -> row i+8g, col r.
// M must be a multiple of 16 (50000 = 3125*16).
// ---------------------------------------------------------------------------
template <int NT>
__global__ __launch_bounds__(128) void gemm_wmma_strip(
    const float* __restrict__ A, const float* __restrict__ W,
    float* __restrict__ C, int M, int K) {
    const int N = NT * 16;
    const int strip = blockIdx.x * 4 + threadIdx.y;       // 4 waves / block
    if (strip >= (M >> 4)) return;                         // whole-wave exit: EXEC stays full
    const int lane = threadIdx.x;
    const int g = lane >> 4;        // 0 or 1
    const int r = lane & 15;

    const float* __restrict__ Arow = A + (size_t)(strip * 16 + r) * K;

    v8f acc[NT] = {};
    for (int k = 0; k < K; k += 4) {
        const int kb = k + 2 * g;
        v2f a;
        a.x = Arow[kb];
        a.y = Arow[kb + 1];
        const float* __restrict__ Wr0 = W + (size_t)kb * N + r;
        const float* __restrict__ Wr1 = Wr0 + N;
#pragma unroll
        for (int t = 0; t < NT; t++) {
            v2f b;
            b.x = Wr0[t * 16];
            b.y = Wr1[t * 16];
            acc[t] = __builtin_amdgcn_wmma_f32_16x16x4_f32(
                /*neg_a=*/false, a, /*neg_b=*/false, b,
                /*c_mod=*/(short)0, acc[t], /*reuse_a=*/false, /*reuse_b=*/false);
        }
    }

#pragma unroll
    for (int t = 0; t < NT; t++) {
        size_t cbase = (size_t)(strip * 16 + 8 * g) * N + t * 16 + r;
#pragma unroll
        for (int i = 0; i < 8; i++) C[cbase + (size_t)i * N] = acc[t][i];
    }
}

// ---------------------------------------------------------------------------
// Edge aggregation: agg[dst] += h[src] * norm[e].
// 4 features per thread: one b128 gather, 4 per-dword L2 float atomics,
// 4x fewer redundant src/dst/norm index loads.
// ---------------------------------------------------------------------------
template <int LOGF>
__global__ void scatter4_kernel(const float4* __restrict__ h4, const int* __restrict__ src,
                                const int* __restrict__ dst, const float* __restrict__ norm,
                                float* __restrict__ agg, int E) {
    const int LOGQ = LOGF - 2;                 // float4 chunks per row
    long long t = (long long)blockIdx.x * blockDim.x + threadIdx.x;
    int e = (int)(t >> LOGQ);
    if (e >= E) return;
    int q = (int)t & ((1 << LOGQ) - 1);
    float nv = norm[e];
    float4 v = h4[((size_t)src[e] << LOGQ) + q];
    float* base = agg + ((size_t)dst[e] << LOGF) + q * 4;
    atomicAdd(base + 0, v.x * nv);
    atomicAdd(base + 1, v.y * nv);
    atomicAdd(base + 2, v.z * nv);
    atomicAdd(base + 3, v.w * nv);
}

// out = agg + h * dinv^2 + bias, optional ReLU (float4 vectorized)
template <int LOGF, bool RELU>
__global__ void finalize4_kernel(const float4* __restrict__ agg4, const float4* __restrict__ h4,
                                 const float* __restrict__ dinv, const float4* __restrict__ bias4,
                                 float4* __restrict__ out4, int Nn) {
    const int LOGQ = LOGF - 2;
    long long t = (long long)blockIdx.x * blockDim.x + threadIdx.x;
    int i = (int)(t >> LOGQ);
    if (i >= Nn) return;
    int q = (int)t & ((1 << LOGQ) - 1);
    float d2 = dinv[i];
    d2 *= d2;
    float4 a = agg4[t], h = h4[t], b = bias4[q], o;
    o.x = a.x + h.x * d2 + b.x;
    o.y = a.y + h.y * d2 + b.y;
    o.z = a.z + h.z * d2 + b.z;
    o.w = a.w + h.w * d2 + b.w;
    if (RELU) {
        o.x = fmaxf(o.x, 0.f); o.y = fmaxf(o.y, 0.f);
        o.z = fmaxf(o.z, 0.f); o.w = fmaxf(o.w, 0.f);
    }
    out4[t] = o;
}

// ---------------------------------------------------------------------------
// Global mean pool over 64 graphs
// ---------------------------------------------------------------------------
template <int LOGF>
__global__ void pool_accum_kernel(const float* __restrict__ h, const int* __restrict__ batch,
                                  float* __restrict__ sums, float* __restrict__ counts, int Nn) {
    long long t = (long long)blockIdx.x * blockDim.x + threadIdx.x;
    int i = (int)(t >> LOGF);
    if (i >= Nn) return;
    int f = (int)t & ((1 << LOGF) - 1);
    int gidx = batch[i];
    atomicAdd(&sums[((size_t)gidx << LOGF) + f], h[t]);
    if (f == 0) atomicAdd(&counts[gidx], 1.0f);
}

__global__ void pool_div_kernel(const float* __restrict__ sums, const float* __restrict__ counts,
                                float* __restrict__ out, int total, int F) {
    int t = blockIdx.x * blockDim.x + threadIdx.x;
    if (t < total) out[t] = sums[t] / fmaxf(counts[t / F], 1.0f);
}

// ---------------------------------------------------------------------------
// Host-side orchestration (graph-capture safe: only async ops on `stream`)
// ---------------------------------------------------------------------------
extern "C" void kernel_launch(void* const* d_in, const int* in_sizes, int n_in,
                              void* d_out, int out_size, void* d_ws, size_t ws_size,
                              hipStream_t stream) {
    const float* x    = (const float*)d_in[0];
    const int*   ei   = (const int*)d_in[1];
    const int*   batch= (const int*)d_in[2];
    const float* W1   = (const float*)d_in[3];
    const float* b1   = (const float*)d_in[4];
    const float* W2   = (const float*)d_in[5];
    const float* b2   = (const float*)d_in[6];
    const float* W3   = (const float*)d_in[7];
    const float* b3   = (const float*)d_in[8];
    float* out = (float*)d_out;

    const int N      = in_sizes[2];          // 50000 nodes
    const int E      = in_sizes[1] / 2;      // 800000 edges
    const int IN_DIM = in_sizes[0] / N;      // 256
    const int HID    = 128;
    const int OUT    = 64;
    const int G      = 64;                   // graphs
    const int* src = ei;
    const int* dst = ei + E;

    // Workspace carve-out (all 256B aligned)
    char* ws = (char*)d_ws;
    size_t off = 0;
    auto carve = [&](size_t bytes) {
        void* p = ws + off;
        off += (bytes + 255) & ~(size_t)255;
        return p;
    };
    float* hbuf  = (float*)carve((size_t)N * HID * sizeof(float)); // x@W result
    float* agg   = (float*)carve((size_t)N * HID * sizeof(float)); // scatter target
    float* hcur  = (float*)carve((size_t)N * HID * sizeof(float)); // activations
    float* dinv  = (float*)carve((size_t)N * sizeof(float));       // deg -> d^-1/2
    float* enorm = (float*)carve((size_t)E * sizeof(float));       // per-edge norm
    float* sums  = (float*)carve((size_t)G * OUT * sizeof(float));
    float* cnts  = (float*)carve((size_t)G * sizeof(float));

    const int TB = 256;
    auto blocks = [](long long n, int tb) { return (unsigned)((n + tb - 1) / tb); };

    // --- degrees and symmetric normalization ---
    hipMemsetAsync(dinv, 0, (size_t)N * sizeof(float), stream);
    deg_kernel<<<blocks(E, TB), TB, 0, stream>>>(dst, dinv, E);
    dinv_kernel<<<blocks(N, TB), TB, 0, stream>>>(dinv, N);
    edge_norm_kernel<<<blocks(E, TB), TB, 0, stream>>>(src, dst, dinv, enorm, E);

    dim3 gblk(32, 4);
    const unsigned ggrid = (unsigned)(((N >> 4) + 3) / 4);   // strips / 4 waves

    // --- Layer 1: [N,256] @ [256,128] ---
    gemm_wmma_strip<8><<<ggrid, gblk, 0, stream>>>(x, W1, hbuf, N, IN_DIM);
    hipMemsetAsync(agg, 0, (size_t)N * HID * sizeof(float), stream);
    scatter4_kernel<7><<<blocks((long long)E * (HID / 4), TB), TB, 0, stream>>>(
        (const float4*)hbuf, src, dst, enorm, agg, E);
    finalize4_kernel<7, true><<<blocks((long long)N * (HID / 4), TB), TB, 0, stream>>>(
        (const float4*)agg, (const float4*)hbuf, dinv, (const float4*)b1, (float4*)hcur, N);

    // --- Layer 2: [N,128] @ [128,128] ---
    gemm_wmma_strip<8><<<ggrid, gblk, 0, stream>>>(hcur, W2, hbuf, N, HID);
    hipMemsetAsync(agg, 0, (size_t)N * HID * sizeof(float), stream);
    scatter4_kernel<7><<<blocks((long long)E * (HID / 4), TB), TB, 0, stream>>>(
        (const float4*)hbuf, src, dst, enorm, agg, E);
    finalize4_kernel<7, true><<<blocks((long long)N * (HID / 4), TB), TB, 0, stream>>>(
        (const float4*)agg, (const float4*)hbuf, dinv, (const float4*)b2, (float4*)hcur, N);

    // --- Layer 3: [N,128] @ [128,64] ---
    gemm_wmma_strip<4><<<ggrid, gblk, 0, stream>>>(hcur, W3, hbuf, N, HID);
    hipMemsetAsync(agg, 0, (size_t)N * OUT * sizeof(float), stream);
    scatter4_kernel<6><<<blocks((long long)E * (OUT / 4), TB), TB, 0, stream>>>(
        (const float4*)hbuf, src, dst, enorm, agg, E);
    finalize4_kernel<6, false><<<blocks((long long)N * (OUT / 4), TB), TB, 0, stream>>>(
        (const float4*)agg, (const float4*)hbuf, dinv, (const float4*)b3, (float4*)hcur, N);

    // --- Global mean pool ---
    hipMemsetAsync(sums, 0, (size_t)G * OUT * sizeof(float), stream);
    hipMemsetAsync(cnts, 0, (size_t)G * sizeof(float), stream);
    pool_accum_kernel<6><<<blocks((long long)N * OUT, TB), TB, 0, stream>>>(hcur, batch, sums, cnts, N);
    pool_div_kernel<<<blocks(G * OUT, TB), TB, 0, stream>>>(sums, cnts, out, G * OUT, OUT);
}